// PointNet2_23012434772463
// MI455X (gfx1250) — compile-verified
//
#include <hip/hip_runtime.h>
#include <math.h>

// ---------------------------------------------------------------------------
// PointNet++ forward for MI455X (gfx1250, wave32).
// All GEMM-shaped work runs through a WMMA f32 16x16x4 GEMM (full fp32
// precision, matching the fp32 reference). Irregular stages (FPS, ball query,
// 3-NN interp, norms, pooling, FC head) are small VALU kernels.
//
// Assumptions (compile-only environment, cannot verify at runtime):
//  * d_in order = setup_inputs() insertion order (see INPUT INDEX MAP below).
//  * ws_size >= ~139 MB (two 68MB GEMM ping-pong buffers + small buffers).
//  * WMMA f32 16x16x4 fragment layouts per CDNA5 ISA 7.12.2:
//      A: lane<16 -> {A[m][k], A[m][k+1]}, lane>=16 -> {A[m][k+2], A[m][k+3]}
//      B (assumed mirror): lane<16 -> {B[k][n], B[k+1][n]}, lane>=16 -> K+2/K+3
//      C/D: vgpr v, lane l -> row = v + 8*(l/16), col = l%16
// ---------------------------------------------------------------------------

#define Bb 8
#define Nn 16384

typedef float v2f __attribute__((ext_vector_type(2)));
typedef float v8f __attribute__((ext_vector_type(8)));

// ---------------------------------------------------------------------------
// Generic GEMM: Y[r,o] = sum_c X[r,c] * W[o,c] + bias[o]
// One wave per 16x16 output tile; K loop in steps of 4 via v_wmma_f32_16x16x4_f32.
//
// OOB A-rows / B-cols only contribute to output elements that are never
// stored, so fragment loads are UNGUARDED (row pointers clamped for memory
// safety); only the K tail (C%4) is zero-padded, and only stores are guarded.
// Main loop is software-pipelined: next fragment loads issue while the
// current WMMA executes. Waves within a block share one W tile (rt fastest).
// ---------------------------------------------------------------------------
__global__ void gemm_bias_wmma(const float* __restrict__ X, const float* __restrict__ W,
                               const float* __restrict__ bias, float* __restrict__ Y,
                               int R, int C, int O) {
  int wid  = (int)(((long long)blockIdx.x * blockDim.x + threadIdx.x) >> 5);
  int lane = (int)(threadIdx.x & 31);
  int rtiles = (R + 15) >> 4;
  int otiles = (O + 15) >> 4;
  if (wid >= rtiles * otiles) return;   // wave-uniform exit
  int rt = wid % rtiles;                // row tile fastest: block shares W tile
  int ot = wid / rtiles;
  int khalf = lane >> 4;                // 0: lanes 0-15, 1: lanes 16-31
  int lid   = lane & 15;
  int arow = rt * 16 + lid;
  int bcol = ot * 16 + lid;
  bool bcol_ok = bcol < O;
  const float* xrow = X + (size_t)(arow < R ? arow : 0) * C + 2 * khalf;
  const float* wrow = W + (size_t)(bcol_ok ? bcol : 0) * C + 2 * khalf;
  v8f acc = {0.f, 0.f, 0.f, 0.f, 0.f, 0.f, 0.f, 0.f};

  int Cmain = C & ~3;
  if (Cmain > 0) {
    v2f a, b;
    a.x = xrow[0]; a.y = xrow[1];
    b.x = wrow[0]; b.y = wrow[1];
#pragma unroll 2
    for (int k = 4; k < Cmain; k += 4) {
      v2f an, bn;
      an.x = xrow[k];     an.y = xrow[k + 1];
      bn.x = wrow[k];     bn.y = wrow[k + 1];
      // 8 args: (neg_a, A, neg_b, B, c_mod, C, reuse_a, reuse_b)
      acc = __builtin_amdgcn_wmma_f32_16x16x4_f32(false, a, false, b, (short)0, acc,
                                                  false, false);
      a = an; b = bn;
    }
    acc = __builtin_amdgcn_wmma_f32_16x16x4_f32(false, a, false, b, (short)0, acc,
                                                false, false);
  }
  if (Cmain < C) {                       // K tail: zero-pad past C
    int k0 = Cmain + 2 * khalf;
    v2f a, b;
    a.x = (k0     < C) ? xrow[Cmain]     : 0.f;
    a.y = (k0 + 1 < C) ? xrow[Cmain + 1] : 0.f;
    b.x = (k0     < C) ? wrow[Cmain]     : 0.f;
    b.y = (k0 + 1 < C) ? wrow[Cmain + 1] : 0.f;
    acc = __builtin_amdgcn_wmma_f32_16x16x4_f32(false, a, false, b, (short)0, acc,
                                                false, false);
  }

  if (bcol_ok) {
    float bv = bias ? bias[bcol] : 0.f;
#pragma unroll
    for (int v = 0; v < 8; ++v) {
      int row = rt * 16 + v + 8 * khalf;
      if (row < R) Y[(size_t)row * O + bcol] = acc[v] + bv;
    }
  }
}

// ---------------------------------------------------------------------------
// Per-output-channel mean / rsqrt(var+eps) over all R rows (BatchNorm stats).
// One block (256 threads) per channel.
// ---------------------------------------------------------------------------
__global__ void colstats_kernel(const float* __restrict__ Y, float* __restrict__ mean,
                                float* __restrict__ rstd, int R, int O, float eps) {
  int o = blockIdx.x;
  float s = 0.f, s2 = 0.f;
  for (int r = threadIdx.x; r < R; r += blockDim.x) {
    float v = Y[(size_t)r * O + o];
    s += v; s2 += v * v;
  }
  __shared__ float sh[256], sh2[256];
  int t = threadIdx.x;
  sh[t] = s; sh2[t] = s2;
  __syncthreads();
  for (int st = 128; st > 0; st >>= 1) {
    if (t < st) { sh[t] += sh[t + st]; sh2[t] += sh2[t + st]; }
    __syncthreads();
  }
  if (t == 0) {
    float m = sh[0] / (float)R;
    float var = sh2[0] / (float)R - m * m;
    mean[o] = m;
    rstd[o] = rsqrtf(var + eps);
  }
}

__global__ void bn_relu_kernel(float* __restrict__ Y, const float* __restrict__ mean,
                               const float* __restrict__ rstd, const float* __restrict__ g,
                               const float* __restrict__ beta, long long total, int O) {
  long long i = (long long)blockIdx.x * blockDim.x + threadIdx.x;
  if (i >= total) return;
  int o = (int)(i % O);
  float y = (Y[i] - mean[o]) * rstd[o] * g[o] + beta[o];
  Y[i] = fmaxf(y, 0.f);
}

// max over middle axis: in (G, S, O) -> out (G, O)
__global__ void maxpool_mid_kernel(const float* __restrict__ in, float* __restrict__ out,
                                   int G, int S, int O) {
  long long i = (long long)blockIdx.x * blockDim.x + threadIdx.x;
  if (i >= (long long)G * O) return;
  int g = (int)(i / O), o = (int)(i % O);
  float m = -3.4e38f;
  for (int k = 0; k < S; ++k) m = fmaxf(m, in[((size_t)g * S + k) * O + o]);
  out[(size_t)g * O + o] = m;
}

// ---------------------------------------------------------------------------
// Farthest point sampling. One block per batch; dist kept in global ws.
// idx[b, it] = incoming farthest index (matches reference scan semantics).
// ---------------------------------------------------------------------------
__global__ void fps_kernel(const float* __restrict__ xyz, int n, int npoint,
                           int* __restrict__ idx_out, float* __restrict__ dist_ws) {
  int b = blockIdx.x;
  const float* p = xyz + (size_t)b * n * 3;
  float* dist = dist_ws + (size_t)b * n;
  int t = threadIdx.x, T = blockDim.x;
  __shared__ float smax[256];
  __shared__ int   simax[256];
  __shared__ int   far_s;
  for (int i = t; i < n; i += T) dist[i] = 1e10f;
  if (t == 0) far_s = 0;
  __syncthreads();
  for (int it = 0; it < npoint; ++it) {
    int far = far_s;
    if (t == 0) idx_out[b * npoint + it] = far;
    float cx = p[far * 3 + 0], cy = p[far * 3 + 1], cz = p[far * 3 + 2];
    float best = -1.f; int bi = 0;
    for (int i = t; i < n; i += T) {
      float dx = p[i * 3 + 0] - cx, dy = p[i * 3 + 1] - cy, dz = p[i * 3 + 2] - cz;
      float d = dx * dx + dy * dy + dz * dz;
      float dm = fminf(dist[i], d);
      dist[i] = dm;
      if (dm > best) { best = dm; bi = i; }
    }
    smax[t] = best; simax[t] = bi;
    __syncthreads();
    for (int st = 128; st > 0; st >>= 1) {
      if (t < st) {
        if (smax[t + st] > smax[t] ||
            (smax[t + st] == smax[t] && simax[t + st] < simax[t])) {
          smax[t] = smax[t + st]; simax[t] = simax[t + st];
        }
      }
      __syncthreads();
    }
    if (t == 0) far_s = simax[0];
    __syncthreads();
  }
}

__global__ void gather3_kernel(const float* __restrict__ xyz, const int* __restrict__ idx,
                               float* __restrict__ out, int n, int npoint) {
  int i = blockIdx.x * blockDim.x + threadIdx.x;
  if (i >= Bb * npoint) return;
  int b = i / npoint;
  int gi = idx[i];
  out[i * 3 + 0] = xyz[((size_t)b * n + gi) * 3 + 0];
  out[i * 3 + 1] = xyz[((size_t)b * n + gi) * 3 + 1];
  out[i * 3 + 2] = xyz[((size_t)b * n + gi) * 3 + 2];
}

// ---------------------------------------------------------------------------
// Ball query: first nsample indices (ascending) with d^2 <= r^2, padded with
// the first hit. One wave32 per query center; ballot-based ordered append.
// ---------------------------------------------------------------------------
__global__ void query_ball_kernel(const float* __restrict__ xyz, const float* __restrict__ new_xyz,
                                  int n, int npoint, int nsample, float r2,
                                  int* __restrict__ gidx) {
  int wid  = (int)(((long long)blockIdx.x * blockDim.x + threadIdx.x) >> 5);
  int lane = (int)(threadIdx.x & 31);
  if (wid >= Bb * npoint) return;           // wave-uniform
  int b = wid / npoint;
  float cx = new_xyz[(size_t)wid * 3 + 0];
  float cy = new_xyz[(size_t)wid * 3 + 1];
  float cz = new_xyz[(size_t)wid * 3 + 2];
  const float* p = xyz + (size_t)b * n * 3;
  int* out = gidx + (size_t)wid * nsample;
  int cnt = 0;
  int firstIdx = -1;
  for (int j0 = 0; j0 < n && cnt < nsample; j0 += 32) {
    int j = j0 + lane;
    bool inr = false;
    if (j < n) {
      float dx = p[j * 3 + 0] - cx, dy = p[j * 3 + 1] - cy, dz = p[j * 3 + 2] - cz;
      inr = (dx * dx + dy * dy + dz * dz) <= r2;
    }
    unsigned mask = (unsigned)__ballot(inr);    // wave32: low 32 bits
    if (firstIdx < 0 && mask != 0u) firstIdx = j0 + (__ffs((int)mask) - 1);
    if (inr) {
      int rank = __popc(mask & ((1u << lane) - 1u));
      int pos = cnt + rank;
      if (pos < nsample) out[pos] = j;
    }
    cnt += __popc(mask);
  }
  if (firstIdx < 0) firstIdx = 0;
  int filled = cnt < nsample ? cnt : nsample;
  for (int pos = filled + lane; pos < nsample; pos += 32) out[pos] = firstIdx;
}

// ---------------------------------------------------------------------------
// SA grouping: X[row, :] = [xyz[gi]-center (3), pts[gi] (Cp)]
// row = (b*npoint + s)*nsample + k
// ---------------------------------------------------------------------------
__global__ void group_kernel(const float* __restrict__ xyz, const float* __restrict__ pts,
                             const float* __restrict__ new_xyz, const int* __restrict__ gidx,
                             float* __restrict__ X, int n, int npoint, int nsample, int Cp) {
  int idx = blockIdx.x * blockDim.x + threadIdx.x;
  int total = Bb * npoint * nsample;
  if (idx >= total) return;
  int s = (idx / nsample) % npoint;
  int b = idx / (nsample * npoint);
  int gi = gidx[idx];
  const float* pj = xyz + ((size_t)b * n + gi) * 3;
  const float* cj = new_xyz + ((size_t)b * npoint + s) * 3;
  float* xr = X + (size_t)idx * (3 + Cp);
  xr[0] = pj[0] - cj[0];
  xr[1] = pj[1] - cj[1];
  xr[2] = pj[2] - cj[2];
  for (int c = 0; c < Cp; ++c) xr[3 + c] = pts[((size_t)b * n + gi) * Cp + c];
}

// X[r, :] = [A[r, :C1], Bp[r/bgroup, :C2]]   (bgroup=1: row-aligned concat;
// bgroup=n1: per-batch broadcast as in set_abstraction_all / fp3)
__global__ void concat2_kernel(const float* __restrict__ A, const float* __restrict__ Bp,
                               float* __restrict__ X, long long rows, int C1, int C2, int bgroup) {
  long long i = (long long)blockIdx.x * blockDim.x + threadIdx.x;
  long long total = rows * (C1 + C2);
  if (i >= total) return;
  long long r = i / (C1 + C2);
  int c = (int)(i % (C1 + C2));
  X[i] = (c < C1) ? A[r * C1 + c] : Bp[(r / bgroup) * (long long)C2 + (c - C1)];
}

// ---------------------------------------------------------------------------
// Feature propagation: 3-NN inverse-distance interp + concat with pts1.
// X[row, :] = [pts1[row, :C1], sum_j w_j * pts2[b, nn_j, :C2]]
// ---------------------------------------------------------------------------
__global__ void interp_concat_kernel(const float* __restrict__ xyz1, const float* __restrict__ xyz2,
                                     const float* __restrict__ pts1, const float* __restrict__ pts2,
                                     float* __restrict__ X, int n1, int n2, int C1, int C2) {
  int idx = blockIdx.x * blockDim.x + threadIdx.x;
  if (idx >= Bb * n1) return;
  int b = idx / n1;
  float px = xyz1[(size_t)idx * 3 + 0];
  float py = xyz1[(size_t)idx * 3 + 1];
  float pz = xyz1[(size_t)idx * 3 + 2];
  const float* q = xyz2 + (size_t)b * n2 * 3;
  float d0 = 1e30f, d1 = 1e30f, d2 = 1e30f;
  int i0 = 0, i1 = 0, i2 = 0;
  for (int j = 0; j < n2; ++j) {
    float dx = q[j * 3 + 0] - px, dy = q[j * 3 + 1] - py, dz = q[j * 3 + 2] - pz;
    float d = dx * dx + dy * dy + dz * dz;
    if (d < d0)      { d2 = d1; i2 = i1; d1 = d0; i1 = i0; d0 = d; i0 = j; }
    else if (d < d1) { d2 = d1; i2 = i1; d1 = d;  i1 = j; }
    else if (d < d2) { d2 = d;  i2 = j; }
  }
  float w0 = 1.f / (d0 + 1e-8f), w1 = 1.f / (d1 + 1e-8f), w2 = 1.f / (d2 + 1e-8f);
  float ws = w0 + w1 + w2;
  w0 /= ws; w1 /= ws; w2 /= ws;
  float* xr = X + (size_t)idx * (C1 + C2);
  for (int c = 0; c < C1; ++c) xr[c] = pts1[(size_t)idx * C1 + c];
  const float* p0 = pts2 + ((size_t)b * n2 + i0) * C2;
  const float* p1 = pts2 + ((size_t)b * n2 + i1) * C2;
  const float* p2 = pts2 + ((size_t)b * n2 + i2) * C2;
  for (int c = 0; c < C2; ++c) xr[C1 + c] = w0 * p0[c] + w1 * p1[c] + w2 * p2[c];
}

// conv1: T[(b*8+o)*N + i] = dot(l0[b,i,:128], Wc[o,:]) + bc[o]  (transposed layout)
__global__ void conv1_kernel(const float* __restrict__ l0, const float* __restrict__ Wc,
                             const float* __restrict__ bc, float* __restrict__ T) {
  int idx = blockIdx.x * blockDim.x + threadIdx.x;
  if (idx >= Bb * Nn) return;
  int b = idx / Nn, i = idx % Nn;
  const float* x = l0 + (size_t)idx * 128;
#pragma unroll
  for (int o = 0; o < 8; ++o) {
    float s = bc[o];
    const float* w = Wc + o * 128;
    for (int c = 0; c < 128; ++c) s += x[c] * w[c];
    T[((size_t)b * 8 + o) * Nn + i] = s;
  }
}

// per-row (contiguous length L) mean / rstd for instance-norm
__global__ void rowstats_kernel(const float* __restrict__ T, float* __restrict__ mean,
                                float* __restrict__ rstd, int L, float eps) {
  int g = blockIdx.x;
  const float* row = T + (size_t)g * L;
  float s = 0.f, s2 = 0.f;
  for (int i = threadIdx.x; i < L; i += blockDim.x) {
    float v = row[i];
    s += v; s2 += v * v;
  }
  __shared__ float sh[256], sh2[256];
  int t = threadIdx.x;
  sh[t] = s; sh2[t] = s2;
  __syncthreads();
  for (int st = 128; st > 0; st >>= 1) {
    if (t < st) { sh[t] += sh[t + st]; sh2[t] += sh2[t + st]; }
    __syncthreads();
  }
  if (t == 0) {
    float m = sh[0] / (float)L;
    float var = sh2[0] / (float)L - m * m;
    mean[g] = m;
    rstd[g] = rsqrtf(var + eps);
  }
}

__global__ void ppf_kernel(const float* __restrict__ T, const float* __restrict__ mean,
                           const float* __restrict__ rstd, float* __restrict__ out,
                           long long total, int L) {
  long long i = (long long)blockIdx.x * blockDim.x + threadIdx.x;
  if (i >= total) return;
  int g = (int)(i / L);
  out[i] = fmaxf((T[i] - mean[g]) * rstd[g], 0.f);
}

// ---------------------------------------------------------------------------
// FC head: x0(8,1024) -> fc1(512)+IN+relu -> fc2(256)+IN+relu -> fc3(256).
// One block (256 threads) per batch row, everything in LDS.
// ---------------------------------------------------------------------------
__global__ void fc_head_kernel(const float* __restrict__ x0,
                               const float* __restrict__ W1, const float* __restrict__ b1,
                               const float* __restrict__ W2, const float* __restrict__ b2,
                               const float* __restrict__ W3, const float* __restrict__ b3,
                               float* __restrict__ out) {
  __shared__ float xs[1024];
  __shared__ float y1[512];
  __shared__ float y2[256];
  __shared__ float red[256];
  int b = blockIdx.x, t = threadIdx.x;
  for (int i = t; i < 1024; i += 256) xs[i] = x0[b * 1024 + i];
  __syncthreads();
  for (int o = t; o < 512; o += 256) {
    float s = b1[o];
    const float* w = W1 + (size_t)o * 1024;
    for (int c = 0; c < 1024; ++c) s += xs[c] * w[c];
    y1[o] = s;
  }
  __syncthreads();
  // inst_norm over 512 + relu
  float ps = 0.f, ps2 = 0.f;
  for (int o = t; o < 512; o += 256) { float v = y1[o]; ps += v; ps2 += v * v; }
  red[t] = ps; __syncthreads();
  for (int st = 128; st > 0; st >>= 1) { if (t < st) red[t] += red[t + st]; __syncthreads(); }
  float m = red[0] / 512.f; __syncthreads();
  red[t] = ps2; __syncthreads();
  for (int st = 128; st > 0; st >>= 1) { if (t < st) red[t] += red[t + st]; __syncthreads(); }
  float rs = rsqrtf(red[0] / 512.f - m * m + 1e-5f); __syncthreads();
  for (int o = t; o < 512; o += 256) y1[o] = fmaxf((y1[o] - m) * rs, 0.f);
  __syncthreads();
  // fc2
  {
    float s = b2[t];
    const float* w = W2 + (size_t)t * 512;
    for (int c = 0; c < 512; ++c) s += y1[c] * w[c];
    y2[t] = s;
  }
  __syncthreads();
  ps = y2[t]; ps2 = ps * ps;
  red[t] = ps; __syncthreads();
  for (int st = 128; st > 0; st >>= 1) { if (t < st) red[t] += red[t + st]; __syncthreads(); }
  m = red[0] / 256.f; __syncthreads();
  red[t] = ps2; __syncthreads();
  for (int st = 128; st > 0; st >>= 1) { if (t < st) red[t] += red[t + st]; __syncthreads(); }
  rs = rsqrtf(red[0] / 256.f - m * m + 1e-5f); __syncthreads();
  y2[t] = fmaxf((y2[t] - m) * rs, 0.f);
  __syncthreads();
  // fc3
  {
    float s = b3[t];
    const float* w = W3 + (size_t)t * 256;
    for (int c = 0; c < 256; ++c) s += y2[c] * w[c];
    out[b * 256 + t] = s;
  }
}

// ---------------------------------------------------------------------------
// Host orchestration
// ---------------------------------------------------------------------------
// INPUT INDEX MAP (setup_inputs() insertion order; each MLP layer: W,b,g,beta)
//  0: xyz (8,16384,3)
//  1..12 : sa1  L0 W(64,3)    L1 W(64,64)   L2 W(128,64)
// 13..24 : sa2  L0 W(128,131) L1 W(128,128) L2 W(256,128)
// 25..36 : sa3  L0 W(256,259) L1 W(512,256) L2 W(1024,512)
// 37..44 : fp3  L0 W(256,1280) L1 W(256,256)
// 45..52 : fp2  L0 W(256,384)  L1 W(128,256)
// 53..64 : fp1  L0 W(128,131)  L1 W(128,128) L2 W(128,128)
// 65,66  : fc1 W(512,1024), b
// 67,68  : fc2 W(256,512),  b
// 69,70  : fc3 W(256,256),  b
// 71,72  : conv1 W(8,128),  b

extern "C" void kernel_launch(void* const* d_in, const int* in_sizes, int n_in,
                              void* d_out, int out_size, void* d_ws, size_t ws_size,
                              hipStream_t stream) {
  (void)in_sizes; (void)n_in; (void)out_size; (void)ws_size;
  auto P = [&](int i) { return (const float*)d_in[i]; };
  const float* xyz = P(0);

  // ---- workspace layout ----
  char* w = (char*)d_ws;
  size_t off = 0;
  auto take = [&](size_t bytes) -> void* {
    void* p = w + off;
    off = (off + bytes + 255) & ~(size_t)255;
    return p;
  };
  float* bigA    = (float*)take((size_t)131072 * 131 * 4);  // GEMM ping
  float* bigB    = (float*)take((size_t)131072 * 128 * 4);  // GEMM pong
  float* dist    = (float*)take((size_t)Bb * Nn * 4);
  int*   idx1    = (int*)  take((size_t)Bb * 128 * 4);
  int*   idx2    = (int*)  take((size_t)Bb * 32 * 4);
  float* l1_xyz  = (float*)take((size_t)Bb * 128 * 3 * 4);
  float* l2_xyz  = (float*)take((size_t)Bb * 32 * 3 * 4);
  int*   gidx1   = (int*)  take((size_t)Bb * 128 * 32 * 4);
  int*   gidx2   = (int*)  take((size_t)Bb * 32 * 64 * 4);
  float* l1_pts  = (float*)take((size_t)Bb * 128 * 128 * 4);
  float* l2_pts  = (float*)take((size_t)Bb * 32 * 256 * 4);
  float* l3_pts  = (float*)take((size_t)Bb * 1024 * 4);
  float* fp3_out = (float*)take((size_t)Bb * 32 * 256 * 4);
  float* fp2_out = (float*)take((size_t)Bb * 128 * 128 * 4);
  float* mean    = (float*)take(1024 * 4);
  float* rstd    = (float*)take(1024 * 4);
  float* mean64  = (float*)take(64 * 4);
  float* rstd64  = (float*)take(64 * 4);

  float* out_x   = (float*)d_out;            // (8,256)
  float* out_ppf = out_x + Bb * 256;         // (8,8,16384)

  auto cdiv = [](long long a, long long b) { return (unsigned)((a + b - 1) / b); };

  // one MLP layer: GEMM + batch stats over all rows + affine-BN + relu
  auto run_layer = [&](const float* X, float* Y, int R, int C, int O, int widx) {
    long long waves = (long long)((R + 15) / 16) * ((O + 15) / 16);
    gemm_bias_wmma<<<cdiv(waves, 8), 256, 0, stream>>>(X, P(widx), P(widx + 1), Y, R, C, O);
    colstats_kernel<<<O, 256, 0, stream>>>(Y, mean, rstd, R, O, 1e-5f);
    long long total = (long long)R * O;
    bn_relu_kernel<<<cdiv(total, 256), 256, 0, stream>>>(Y, mean, rstd, P(widx + 2),
                                                         P(widx + 3), total, O);
  };

  // ================= SA1: npoint=128, r=0.2, nsample=32 =================
  fps_kernel<<<Bb, 256, 0, stream>>>(xyz, Nn, 128, idx1, dist);
  gather3_kernel<<<cdiv(Bb * 128, 256), 256, 0, stream>>>(xyz, idx1, l1_xyz, Nn, 128);
  query_ball_kernel<<<cdiv((long long)Bb * 128, 8), 256, 0, stream>>>(
      xyz, l1_xyz, Nn, 128, 32, 0.04f, gidx1);
  group_kernel<<<cdiv((long long)Bb * 128 * 32, 256), 256, 0, stream>>>(
      xyz, nullptr, l1_xyz, gidx1, bigA, Nn, 128, 32, 0);
  run_layer(bigA, bigB, 32768, 3, 64, 1);
  run_layer(bigB, bigA, 32768, 64, 64, 5);
  run_layer(bigA, bigB, 32768, 64, 128, 9);
  maxpool_mid_kernel<<<cdiv((long long)1024 * 128, 256), 256, 0, stream>>>(
      bigB, l1_pts, 1024, 32, 128);

  // ================= SA2: npoint=32, r=0.4, nsample=64 ==================
  fps_kernel<<<Bb, 256, 0, stream>>>(l1_xyz, 128, 32, idx2, dist);
  gather3_kernel<<<1, 256, 0, stream>>>(l1_xyz, idx2, l2_xyz, 128, 32);
  query_ball_kernel<<<cdiv((long long)Bb * 32, 8), 256, 0, stream>>>(
      l1_xyz, l2_xyz, 128, 32, 64, 0.16f, gidx2);
  group_kernel<<<cdiv((long long)Bb * 32 * 64, 256), 256, 0, stream>>>(
      l1_xyz, l1_pts, l2_xyz, gidx2, bigA, 128, 32, 64, 128);
  run_layer(bigA, bigB, 16384, 131, 128, 13);
  run_layer(bigB, bigA, 16384, 128, 128, 17);
  run_layer(bigA, bigB, 16384, 128, 256, 21);
  maxpool_mid_kernel<<<cdiv((long long)256 * 256, 256), 256, 0, stream>>>(
      bigB, l2_pts, 256, 64, 256);

  // ================= SA3 (group-all): rows = 8*32, C = 3+256 ============
  concat2_kernel<<<cdiv((long long)256 * 259, 256), 256, 0, stream>>>(
      l2_xyz, l2_pts, bigA, 256, 3, 256, 1);
  run_layer(bigA, bigB, 256, 259, 256, 25);
  run_layer(bigB, bigA, 256, 256, 512, 29);
  run_layer(bigA, bigB, 256, 512, 1024, 33);
  maxpool_mid_kernel<<<cdiv((long long)Bb * 1024, 256), 256, 0, stream>>>(
      bigB, l3_pts, Bb, 32, 1024);

  // ================= FP3: broadcast l3_pts, concat with l2_pts ==========
  concat2_kernel<<<cdiv((long long)256 * 1280, 256), 256, 0, stream>>>(
      l2_pts, l3_pts, bigA, 256, 256, 1024, 32);
  run_layer(bigA, bigB, 256, 1280, 256, 37);
  run_layer(bigB, fp3_out, 256, 256, 256, 41);

  // ================= FP2: interp l2->l1 (3-NN), concat l1_pts ===========
  interp_concat_kernel<<<cdiv((long long)Bb * 128, 256), 256, 0, stream>>>(
      l1_xyz, l2_xyz, l1_pts, fp3_out, bigA, 128, 32, 128, 256);
  run_layer(bigA, bigB, 1024, 384, 256, 45);
  run_layer(bigB, fp2_out, 1024, 256, 128, 49);

  // ================= FP1: interp l1->l0 (3-NN), concat xyz ==============
  interp_concat_kernel<<<cdiv((long long)Bb * Nn, 256), 256, 0, stream>>>(
      xyz, l1_xyz, xyz, fp2_out, bigA, Nn, 128, 3, 128);
  run_layer(bigA, bigB, 131072, 131, 128, 53);
  run_layer(bigB, bigA, 131072, 128, 128, 57);
  run_layer(bigA, bigB, 131072, 128, 128, 61);   // l0_pts -> bigB

  // ================= conv1 + instance-norm + relu -> ppf ================
  conv1_kernel<<<cdiv((long long)Bb * Nn, 256), 256, 0, stream>>>(bigB, P(71), P(72), bigA);
  rowstats_kernel<<<Bb * 8, 256, 0, stream>>>(bigA, mean64, rstd64, Nn, 1e-5f);
  ppf_kernel<<<cdiv((long long)Bb * 8 * Nn, 256), 256, 0, stream>>>(
      bigA, mean64, rstd64, out_ppf, (long long)Bb * 8 * Nn, Nn);

  // ================= FC head -> x =======================================
  fc_head_kernel<<<Bb, 256, 0, stream>>>(l3_pts, P(65), P(66), P(67), P(68), P(69), P(70),
                                         out_x);
}